// DNC_44890998178625
// MI455X (gfx1250) — compile-verified
//
#include <hip/hip_runtime.h>
#include <hip/hip_bf16.h>
#include <math.h>

typedef float v2f __attribute__((ext_vector_type(2)));
typedef float v8f __attribute__((ext_vector_type(8)));

namespace {
constexpr int Bn = 16, Sn = 256, INDIM = 264, Hn = 512, Mn = 64, Rn = 4, Nn = 256;
constexpr int XIn = 471, XIPAD = 480, OUTD = 256;
constexpr int WXROWS = INDIM + Rn * Mn;   // 520
constexpr int FOURH  = 4 * Hn;            // 2048

// workspace layout (float offsets)
constexpr size_t OFF_H   = 0;                                   // 2 x B x H  (ping-pong)
constexpr size_t OFF_C   = OFF_H   + 2ull * Bn * Hn;
constexpr size_t OFF_MEM = OFF_C   + (size_t)Bn * Hn;
constexpr size_t OFF_USG = OFF_MEM + (size_t)Bn * Nn * Mn;
constexpr size_t OFF_LNK = OFF_USG + (size_t)Bn * Nn;
constexpr size_t OFF_PRE = OFF_LNK + (size_t)Bn * Nn * Nn;
constexpr size_t OFF_WR  = OFF_PRE + (size_t)Bn * Nn;
constexpr size_t OFF_WW  = OFF_WR  + (size_t)Bn * Rn * Nn;
constexpr size_t OFF_RV  = OFF_WW  + (size_t)Bn * Nn;
constexpr size_t OFF_XI  = OFF_RV  + (size_t)Bn * Rn * Mn;
constexpr size_t OFF_YH  = OFF_XI  + (size_t)Bn * XIPAD;
constexpr size_t WS_FLOATS = OFF_YH + (size_t)Bn * OUTD;        // ~1.38M floats (~5.5 MB)
} // namespace

__device__ __forceinline__ float sigmoidf(float x) { return 1.f / (1.f + __expf(-x)); }
__device__ __forceinline__ float oneplusf(float x) {
  // 1 + softplus(x), numerically stable
  return 1.f + fmaxf(x, 0.f) + log1pf(__expf(-fabsf(x)));
}

__device__ __forceinline__ float blk_max(float v, float* red, int tid) {
  red[tid] = v; __syncthreads();
  #pragma unroll
  for (int s = 128; s >= 1; s >>= 1) {
    if (tid < s) red[tid] = fmaxf(red[tid], red[tid + s]);
    __syncthreads();
  }
  float r = red[0]; __syncthreads();
  return r;
}

__device__ __forceinline__ float blk_sum(float v, float* red, int tid) {
  red[tid] = v; __syncthreads();
  #pragma unroll
  for (int s = 128; s >= 1; s >>= 1) {
    if (tid < s) red[tid] += red[tid + s];
    __syncthreads();
  }
  float r = red[0]; __syncthreads();
  return r;
}

// Branchless WMMA macro-step: A pair from `ap`, B columns from `wp` (row
// stride FOURH floats), 4 gate accumulators.
#define GATE_STEP(ap, wp)                                                          \
  do {                                                                             \
    v2f a = *(const v2f*)(ap);                                                     \
    v2f b;                                                                         \
    b.x = (wp)[n_i]; b.y = (wp)[FOURH + n_i];                                      \
    ai = __builtin_amdgcn_wmma_f32_16x16x4_f32(false, a, false, b, (short)0, ai,   \
                                               false, false);                      \
    b.x = (wp)[n_f]; b.y = (wp)[FOURH + n_f];                                      \
    af = __builtin_amdgcn_wmma_f32_16x16x4_f32(false, a, false, b, (short)0, af,   \
                                               false, false);                      \
    b.x = (wp)[n_g]; b.y = (wp)[FOURH + n_g];                                      \
    ag = __builtin_amdgcn_wmma_f32_16x16x4_f32(false, a, false, b, (short)0, ag,   \
                                               false, false);                      \
    b.x = (wp)[n_o]; b.y = (wp)[FOURH + n_o];                                      \
    ao = __builtin_amdgcn_wmma_f32_16x16x4_f32(false, a, false, b, (short)0, ao,   \
                                               false, false);                      \
  } while (0)

// -------------------------------------------------------------------------
// Kernel A: fused LSTM. gates = [x|r_vec|h] @ [Wx;Wh] + b, then c/h update.
// Grid: 8 blocks x 128 thr (4 waves). Wave w of block handles h-column-block
// cb = bid*4+w (16 cols), computing all four 16x16 gate tiles via FP32 WMMA.
// K-loop split into three branchless region loops (x / r_vec / h).
// -------------------------------------------------------------------------
__global__ __launch_bounds__(128) void dnc_lstm_kernel(
    const float* __restrict__ x_all, const float* __restrict__ Wx,
    const float* __restrict__ Wh,   const float* __restrict__ b_lstm,
    const float* __restrict__ hin,  float* __restrict__ hout,
    float* __restrict__ cst, const float* __restrict__ rvec, int t)
{
  const int lane = threadIdx.x & 31;
  const int wave = threadIdx.x >> 5;
  const int cb   = blockIdx.x * 4 + wave;   // 0..31 column block of H
  const int ln   = lane & 15;
  const int ksel = (lane >> 4) << 1;        // 0 or 2

  const int n_i = 0 * Hn + cb * 16 + ln;
  const int n_f = 1 * Hn + cb * 16 + ln;
  const int n_g = 2 * Hn + cb * 16 + ln;
  const int n_o = 3 * Hn + cb * 16 + ln;

  v8f ai = {}, af = {}, ag = {}, ao = {};

  // ---- region 1: x_t rows of Wx (K = 0..263) ----
  {
    const float* ap = x_all + ((size_t)ln * Sn + t) * INDIM + ksel;
    const float* wp = Wx + (size_t)ksel * FOURH;
    #pragma unroll 2
    for (int kb = 0; kb < INDIM; kb += 4) {
      if ((kb & 63) == 0) __builtin_prefetch((const void*)(wp + 64 * FOURH + n_i), 0, 0);
      GATE_STEP(ap, wp);
      ap += 4; wp += 4 * FOURH;
    }
  }
  // ---- region 2: r_vec rows of Wx (K = 264..519) ----
  {
    const float* ap = rvec + (size_t)ln * (Rn * Mn) + ksel;
    const float* wp = Wx + (size_t)(INDIM + ksel) * FOURH;
    #pragma unroll 2
    for (int kb = 0; kb < Rn * Mn; kb += 4) {
      GATE_STEP(ap, wp);
      ap += 4; wp += 4 * FOURH;
    }
  }
  // ---- region 3: h rows of Wh (K = 520..1031) ----
  {
    const float* ap = hin + (size_t)ln * Hn + ksel;
    const float* wp = Wh + (size_t)ksel * FOURH;
    #pragma unroll 2
    for (int kb = 0; kb < Hn; kb += 4) {
      if ((kb & 63) == 0 && kb + 68 < Hn)
        __builtin_prefetch((const void*)(wp + 64 * FOURH + n_i), 0, 0);
      GATE_STEP(ap, wp);
      ap += 4; wp += 4 * FOURH;
    }
  }

  const float bi = b_lstm[n_i], bf = b_lstm[n_f], bg = b_lstm[n_g], bo = b_lstm[n_o];
  const int mbase = (lane >> 4) * 8;
  const int ncol  = cb * 16 + ln;
  #pragma unroll
  for (int v = 0; v < 8; ++v) {
    const int m = mbase + v;
    const float ig = sigmoidf(ai[v] + bi);
    const float fg = sigmoidf(af[v] + bf);
    const float gg = tanhf(ag[v] + bg);
    const float og = sigmoidf(ao[v] + bo);
    const float cold = cst[(size_t)m * Hn + ncol];
    const float cn = fg * cold + ig * gg;
    cst[(size_t)m * Hn + ncol]  = cn;
    hout[(size_t)m * Hn + ncol] = og * tanhf(cn);
  }
}

// -------------------------------------------------------------------------
// Kernel B: xi = h @ W_xi + b_xi (tiles 0..29, cols padded to 480)
//           yh = h @ W_y         (tiles 30..45)
// Out-of-range columns handled with clamped pointer + 0/1 mask (branchless,
// in-bounds).
// -------------------------------------------------------------------------
__global__ __launch_bounds__(128) void dnc_proj_kernel(
    const float* __restrict__ h, const float* __restrict__ Wxi,
    const float* __restrict__ bxi, const float* __restrict__ Wy,
    float* __restrict__ xi_ws, float* __restrict__ yh_ws)
{
  const int lane = threadIdx.x & 31;
  const int wave = threadIdx.x >> 5;
  const int tile = blockIdx.x * 4 + wave;     // 0..47
  if (tile >= 30 + OUTD / 16) return;         // wave-uniform exit, EXEC stays full
  const int ln   = lane & 15;
  const int ksel = (lane >> 4) << 1;
  const bool is_xi = (tile < 30);
  const float* W = is_xi ? Wxi : Wy;
  const int WN   = is_xi ? XIn : OUTD;
  const int n    = (is_xi ? tile * 16 : (tile - 30) * 16) + ln;
  const bool valid = (n < WN);
  const float maskf = valid ? 1.f : 0.f;
  const int nc = valid ? n : (WN - 1);

  v8f acc = {};
  {
    const float* ap = h + (size_t)ln * Hn + ksel;
    const float* wp = W + (size_t)ksel * WN + nc;   // column pointer, row stride WN
    #pragma unroll 2
    for (int kb = 0; kb < Hn; kb += 4) {
      v2f a = *(const v2f*)(ap);
      v2f b;
      b.x = wp[0] * maskf;
      b.y = wp[WN] * maskf;
      acc = __builtin_amdgcn_wmma_f32_16x16x4_f32(false, a, false, b, (short)0, acc,
                                                  false, false);
      ap += 4; wp += 4 * WN;
    }
  }
  const int mbase = (lane >> 4) * 8;
  if (is_xi) {
    if (valid) {
      const float bv = bxi[n];
      #pragma unroll
      for (int v = 0; v < 8; ++v)
        xi_ws[(size_t)(mbase + v) * XIPAD + n] = acc[v] + bv;
    }
  } else {
    #pragma unroll
    for (int v = 0; v < 8; ++v)
      yh_ws[(size_t)(mbase + v) * OUTD + n] = acc[v];
  }
}

// -------------------------------------------------------------------------
// Kernel C: per-batch DNC memory machinery. Grid = 16 blocks x 256 threads.
// -------------------------------------------------------------------------
__global__ __launch_bounds__(256) void dnc_mem_kernel(
    float* __restrict__ mem,  float* __restrict__ usage,
    float* __restrict__ link, float* __restrict__ prec,
    float* __restrict__ w_r,  float* __restrict__ w_w,
    float* __restrict__ r_vec, const float* __restrict__ xi_ws,
    const float* __restrict__ yh_ws, const float* __restrict__ Wr_out,
    const float* __restrict__ b_y, float* __restrict__ out, int t)
{
  const int b   = blockIdx.x;
  const int tid = threadIdx.x;

  __shared__ float s_xi[XIPAD];
  __shared__ float s_u[Nn], s_all[Nn], s_ww[Nn], s_prec[Nn], s_cw[Nn];
  __shared__ float s_wrp[Rn][Nn], s_wrn[Rn][Nn];
  __shared__ float s_fwd[Rn][Nn], s_bwd[Rn][Nn];
  __shared__ float s_sim[Nn], s_red[Nn], s_rv[Rn * Mn];
  __shared__ float s_er[Mn], s_wv[Mn];

  float* memb  = mem  + (size_t)b * Nn * Mn;
  float* linkb = link + (size_t)b * Nn * Nn;

  for (int i = tid; i < XIn; i += 256) s_xi[i] = xi_ws[(size_t)b * XIPAD + i];
  for (int i = tid; i < Rn * Nn; i += 256) (&s_wrp[0][0])[i] = w_r[(size_t)b * Rn * Nn + i];
  s_prec[tid] = prec[(size_t)b * Nn + tid];
  __syncthreads();

  if (tid < Mn) {
    s_er[tid] = sigmoidf(s_xi[325 + tid]);   // erase
    s_wv[tid] = s_xi[389 + tid];             // write vector
  }

  // --- phase 1: retention psi, usage update (uses previous w_r, w_w) ---
  {
    float psi = 1.f;
    #pragma unroll
    for (int r = 0; r < Rn; ++r)
      psi *= 1.f - sigmoidf(s_xi[453 + r]) * s_wrp[r][tid];
    const float u   = usage[(size_t)b * Nn + tid];
    const float wwp = w_w[(size_t)b * Nn + tid];
    s_u[tid] = (u + wwp - u * wwp) * psi;
  }
  __syncthreads();

  // --- phase 2: allocation (stable argsort + exclusive cumprod, O(N^2)) ---
  {
    const float ui = s_u[tid];
    float p = 1.f;
    for (int j = 0; j < Nn; ++j) {
      const float uj = s_u[j];
      if (uj < ui || (uj == ui && j < tid)) p *= uj;
    }
    s_all[tid] = (1.f - ui) * p;
    usage[(size_t)b * Nn + tid] = ui;
  }

  // --- phase 3: write content weights (pre-update memory, cosine sim) ---
  {
    const float* mrow = memb + (size_t)tid * Mn;
    float ss = 0.f, dot = 0.f, kk = 0.f;
    for (int m = 0; m < Mn; ++m) {
      const float mv = mrow[m], kv = s_xi[260 + m];
      ss += mv * mv; dot += mv * kv; kk += kv * kv;
    }
    const float beta_w = oneplusf(s_xi[324]);
    s_sim[tid] = beta_w * dot / ((sqrtf(ss) + 1e-6f) * (sqrtf(kk) + 1e-6f));
  }
  {
    const float mx = blk_max(s_sim[tid], s_red, tid);
    const float ev = __expf(s_sim[tid] - mx);
    const float se = blk_sum(ev, s_red, tid);
    s_cw[tid] = ev / se;
  }

  // --- phase 4: write weights ---
  {
    const float g_a = sigmoidf(s_xi[457]);
    const float g_w = sigmoidf(s_xi[458]);
    const float wwn = g_w * (g_a * s_all[tid] + (1.f - g_a) * s_cw[tid]);
    s_ww[tid] = wwn;
    w_w[(size_t)b * Nn + tid] = wwn;
  }
  __syncthreads();

  // --- phase 5: memory write + new norms + read-key dots ---
  float memnorm_new;
  float dotr[Rn] = {0.f, 0.f, 0.f, 0.f};
  {
    float* mrow = memb + (size_t)tid * Mn;
    const float wwi = s_ww[tid];
    float ss = 0.f;
    for (int m = 0; m < Mn; ++m) {
      const float nv = mrow[m] * (1.f - wwi * s_er[m]) + wwi * s_wv[m];
      mrow[m] = nv;
      ss += nv * nv;
      dotr[0] += nv * s_xi[0 * Mn + m];
      dotr[1] += nv * s_xi[1 * Mn + m];
      dotr[2] += nv * s_xi[2 * Mn + m];
      dotr[3] += nv * s_xi[3 * Mn + m];
    }
    memnorm_new = sqrtf(ss);
  }

  // --- phase 6: link matrix + precedence ---
  const float wwsum = blk_sum(s_ww[tid], s_red, tid);
  {
    const float wwi = s_ww[tid];
    float* lrow = linkb + (size_t)tid * Nn;
    for (int j = 0; j < Nn; ++j) {
      const float nl = (1.f - wwi - s_ww[j]) * lrow[j] + wwi * s_prec[j];
      lrow[j] = (j == tid) ? 0.f : nl;
    }
    prec[(size_t)b * Nn + tid] = (1.f - wwsum) * s_prec[tid] + s_ww[tid];
  }
  __threadfence_block();
  __syncthreads();

  // --- phase 7: forward / backward weights (link @ w_r_prev) ---
  {
    float f0 = 0, f1 = 0, f2 = 0, f3 = 0, g0 = 0, g1 = 0, g2 = 0, g3 = 0;
    for (int j = 0; j < Nn; ++j) {
      const float lij = linkb[(size_t)tid * Nn + j];
      const float lji = linkb[(size_t)j * Nn + tid];
      const float w0 = s_wrp[0][j], w1 = s_wrp[1][j], w2 = s_wrp[2][j], w3 = s_wrp[3][j];
      f0 += lij * w0; f1 += lij * w1; f2 += lij * w2; f3 += lij * w3;
      g0 += lji * w0; g1 += lji * w1; g2 += lji * w2; g3 += lji * w3;
    }
    s_fwd[0][tid] = f0; s_fwd[1][tid] = f1; s_fwd[2][tid] = f2; s_fwd[3][tid] = f3;
    s_bwd[0][tid] = g0; s_bwd[1][tid] = g1; s_bwd[2][tid] = g2; s_bwd[3][tid] = g3;
  }
  __syncthreads();

  // --- phase 8: read content weights + new read weights ---
  for (int r = 0; r < Rn; ++r) {
    float kk = 0.f;
    for (int m = 0; m < Mn; ++m) { const float kv = s_xi[r * Mn + m]; kk += kv * kv; }
    const float beta = oneplusf(s_xi[256 + r]);
    const float sim  = beta * dotr[r] / ((memnorm_new + 1e-6f) * (sqrtf(kk) + 1e-6f));
    const float mxr = blk_max(sim, s_red, tid);
    const float er_ = __expf(sim - mxr);
    const float ser = blk_sum(er_, s_red, tid);
    const float cr  = er_ / ser;
    const float p0 = s_xi[459 + r * 3 + 0], p1 = s_xi[459 + r * 3 + 1], p2 = s_xi[459 + r * 3 + 2];
    const float pm = fmaxf(p0, fmaxf(p1, p2));
    const float e0 = __expf(p0 - pm), e1 = __expf(p1 - pm), e2 = __expf(p2 - pm);
    const float wrn = (e0 * s_bwd[r][tid] + e1 * cr + e2 * s_fwd[r][tid]) / (e0 + e1 + e2);
    s_wrn[r][tid] = wrn;
    w_r[(size_t)b * Rn * Nn + r * Nn + tid] = wrn;
  }
  __syncthreads();

  // --- phase 9: read vectors r_vec[r][m] = sum_n w_r[r][n] * mem[n][m] ---
  {
    const int r = tid >> 6, m = tid & (Mn - 1);
    float acc = 0.f;
    for (int n = 0; n < Nn; ++n) acc += s_wrn[r][n] * memb[(size_t)n * Mn + m];
    s_rv[tid] = acc;
    r_vec[(size_t)b * Rn * Mn + tid] = acc;
  }
  __syncthreads();

  // --- phase 10: output y = h@W_y + r_vec@W_r + b_y ---
  {
    float acc = yh_ws[(size_t)b * OUTD + tid] + b_y[tid];
    for (int rm = 0; rm < Rn * Mn; ++rm)
      acc += s_rv[rm] * Wr_out[(size_t)rm * OUTD + tid];
    out[((size_t)b * Sn + t) * OUTD + tid] = acc;
  }
}

__global__ void dnc_zero_kernel(float* __restrict__ p, size_t n) {
  size_t i = (size_t)blockIdx.x * blockDim.x + threadIdx.x;
  const size_t stride = (size_t)gridDim.x * blockDim.x;
  for (; i < n; i += stride) p[i] = 0.f;
}

extern "C" void kernel_launch(void* const* d_in, const int* in_sizes, int n_in,
                              void* d_out, int out_size, void* d_ws, size_t ws_size,
                              hipStream_t stream) {
  (void)in_sizes; (void)n_in; (void)out_size; (void)ws_size;
  const float* x     = (const float*)d_in[0];
  const float* Wx    = (const float*)d_in[1];
  const float* Wh    = (const float*)d_in[2];
  const float* blstm = (const float*)d_in[3];
  const float* Wxi   = (const float*)d_in[4];
  const float* bxi   = (const float*)d_in[5];
  const float* Wy    = (const float*)d_in[6];
  const float* Wr    = (const float*)d_in[7];
  const float* by    = (const float*)d_in[8];
  float* out = (float*)d_out;
  float* ws  = (float*)d_ws;

  float* h   = ws + OFF_H;
  float* c   = ws + OFF_C;
  float* mem = ws + OFF_MEM;
  float* usg = ws + OFF_USG;
  float* lnk = ws + OFF_LNK;
  float* pre = ws + OFF_PRE;
  float* wr  = ws + OFF_WR;
  float* ww  = ws + OFF_WW;
  float* rv  = ws + OFF_RV;
  float* xi  = ws + OFF_XI;
  float* yh  = ws + OFF_YH;

  // deterministic zero-init of all recurrent state each call
  dnc_zero_kernel<<<512, 256, 0, stream>>>(ws, WS_FLOATS);

  for (int t = 0; t < Sn; ++t) {
    const float* hin = h + (size_t)(t & 1) * (Bn * Hn);
    float* hout      = h + (size_t)((t + 1) & 1) * (Bn * Hn);
    dnc_lstm_kernel<<<8, 128, 0, stream>>>(x, Wx, Wh, blstm, hin, hout, c, rv, t);
    dnc_proj_kernel<<<12, 128, 0, stream>>>(hout, Wxi, bxi, Wy, xi, yh);
    dnc_mem_kernel<<<Bn, 256, 0, stream>>>(mem, usg, lnk, pre, wr, ww, rv, xi, yh, Wr, by, out, t);
  }
}